// DASMIL_end_4758823764288
// MI455X (gfx1250) — compile-verified
//
#include <hip/hip_runtime.h>
#include <math.h>

#define N_NODES 1024
#define N_EDGES 32768
#define FEAT    384
#define HID     64

typedef __attribute__((ext_vector_type(2))) float v2f;
typedef __attribute__((ext_vector_type(8))) float v8f;

// ---------------------------------------------------------------------------
// Kernel 1: q = X@Wq, k = X@Wk, v = X@Wv   (blockIdx.z selects matrix)
// One wave32 per 16x16 output tile, V_WMMA_F32_16X16X4_F32, K-loop over 384.
// ---------------------------------------------------------------------------
__global__ __launch_bounds__(32) void qkv_kernel(
    const float* __restrict__ X,
    const float* __restrict__ Wk, const float* __restrict__ Wq, const float* __restrict__ Wv,
    float* __restrict__ q, float* __restrict__ k, float* __restrict__ v)
{
    const float* W; float* O;
    if (blockIdx.z == 0)      { W = Wq; O = q; }
    else if (blockIdx.z == 1) { W = Wk; O = k; }
    else                      { W = Wv; O = v; }

    const int i    = blockIdx.y * 16;          // output row tile
    const int j    = blockIdx.x * 16;          // output col tile
    const int lane = threadIdx.x & 31;
    const int lm   = lane & 15;
    const int koff = (lane >> 4) * 2;          // 0 for lanes 0-15, 2 for lanes 16-31
    const bool hi  = (koff != 0);

    v8f c = {};
    for (int kk = 0; kk < FEAT; kk += 4) {
        // A operand: rows i..i+15, K slice kk..kk+3 (contiguous -> b128 load)
        float4 af = *(const float4*)&X[(i + lm) * FEAT + kk];
        v2f a, b;
        a.x = hi ? af.z : af.x;
        a.y = hi ? af.w : af.y;
        // B operand: W rows kk+koff, kk+koff+1, col j+lm (stride HID)
        b.x = W[(kk + koff)     * HID + j + lm];
        b.y = W[(kk + koff + 1) * HID + j + lm];
        c = __builtin_amdgcn_wmma_f32_16x16x4_f32(false, a, false, b, (short)0, c, false, false);
    }
    const int rbase = i + (lane >> 4) * 8;     // M = r (lanes 0-15), M = r+8 (lanes 16-31)
#pragma unroll
    for (int r = 0; r < 8; ++r)
        O[(rbase + r) * HID + j + lm] = c[r];
}

// ---------------------------------------------------------------------------
// Kernel 2: A = 2 * (q @ k^T)   [term1 + term3 of the reference]
// One wave32 per 16x16 tile of the 1024x1024 score matrix.
// ---------------------------------------------------------------------------
__global__ __launch_bounds__(32) void scores_kernel(
    const float* __restrict__ q, const float* __restrict__ k, float* __restrict__ A)
{
    const int i    = blockIdx.y * 16;
    const int j    = blockIdx.x * 16;
    const int lane = threadIdx.x & 31;
    const int lm   = lane & 15;
    const bool hi  = (lane >> 4) != 0;

    v8f c = {};
    for (int kk = 0; kk < HID; kk += 4) {
        float4 af = *(const float4*)&q[(i + lm) * HID + kk];
        float4 bf = *(const float4*)&k[(j + lm) * HID + kk];   // B = k^T: B[K][n] = k[n][K]
        v2f a, b;
        a.x = hi ? af.z : af.x;  a.y = hi ? af.w : af.y;
        b.x = hi ? bf.z : bf.x;  b.y = hi ? bf.w : bf.y;
        c = __builtin_amdgcn_wmma_f32_16x16x4_f32(false, a, false, b, (short)0, c, false, false);
    }
    const int rbase = i + (lane >> 4) * 8;
#pragma unroll
    for (int r = 0; r < 8; ++r)
        A[(rbase + r) * N_NODES + j + lm] = 2.0f * c[r];
}

// ---------------------------------------------------------------------------
// Kernel 3: per-edge pre-softmax additions.
// A[s,d] += dot(q[s], trunc(rk_e)) + dot(k[d], trunc(rq_e))
// rk_e = w0*Ek[0] + (1-w0)*Ek[1],  w0 = sigmoid((attr-bias)*mult)
// One wave per edge; lane handles h=lane and h=lane+32; shuffle-reduce; one atomic.
// ---------------------------------------------------------------------------
__global__ __launch_bounds__(256) void edge_qk_kernel(
    const int* __restrict__ eidx, const float* __restrict__ attr,
    const float* __restrict__ q, const float* __restrict__ k,
    const float* __restrict__ Ek, const float* __restrict__ Eq,
    const float* __restrict__ bias, const float* __restrict__ mult,
    float* __restrict__ A)
{
    const int wave = blockIdx.x * (blockDim.x >> 5) + (threadIdx.x >> 5);
    const int lane = threadIdx.x & 31;
    if (wave >= N_EDGES) return;

    const int s = eidx[wave];
    const int d = eidx[N_EDGES + wave];
    const float w0 = 1.0f / (1.0f + __expf(-(attr[wave] - bias[0]) * mult[0]));
    const float w1 = 1.0f - w0;

    float acc = 0.0f;
#pragma unroll
    for (int t = 0; t < 2; ++t) {
        const int h = lane + t * 32;
        const float rk = truncf(w0 * Ek[h] + w1 * Ek[HID + h]);
        const float rq = truncf(w0 * Eq[h] + w1 * Eq[HID + h]);
        acc += q[s * HID + h] * rk + k[d * HID + h] * rq;
    }
#pragma unroll
    for (int off = 16; off > 0; off >>= 1)
        acc += __shfl_xor(acc, off, 32);
    if (lane == 0)
        atomicAdd(&A[s * N_NODES + d], acc);
}

// ---------------------------------------------------------------------------
// Kernel 4: row-wise softmax of A with 1/sqrt(384) scale. One 256-thread block per row.
// ---------------------------------------------------------------------------
__global__ __launch_bounds__(256) void softmax_kernel(float* __restrict__ A)
{
    __shared__ float red[256];
    const int row = blockIdx.x;
    const int t   = threadIdx.x;
    const float scale = 0.051031036307982884f;   // 1/sqrt(384)

    float x[4];
    float m = -3.402823466e38f;
#pragma unroll
    for (int c = 0; c < 4; ++c) {
        x[c] = A[row * N_NODES + t + c * 256] * scale;
        m = fmaxf(m, x[c]);
    }
    red[t] = m; __syncthreads();
    for (int s = 128; s > 0; s >>= 1) {
        if (t < s) red[t] = fmaxf(red[t], red[t + s]);
        __syncthreads();
    }
    m = red[0]; __syncthreads();

    float sum = 0.0f;
#pragma unroll
    for (int c = 0; c < 4; ++c) { x[c] = __expf(x[c] - m); sum += x[c]; }
    red[t] = sum; __syncthreads();
    for (int s = 128; s > 0; s >>= 1) {
        if (t < s) red[t] += red[t + s];
        __syncthreads();
    }
    const float inv = 1.0f / red[0];
#pragma unroll
    for (int c = 0; c < 4; ++c)
        A[row * N_NODES + t + c * 256] = x[c] * inv;
}

// ---------------------------------------------------------------------------
// Kernel 5: M = A @ v  -> d_out.  One wave32 per 16x16 tile, K-loop over 1024.
// ---------------------------------------------------------------------------
__global__ __launch_bounds__(32) void av_kernel(
    const float* __restrict__ A, const float* __restrict__ v, float* __restrict__ M)
{
    const int i    = blockIdx.y * 16;
    const int j    = blockIdx.x * 16;
    const int lane = threadIdx.x & 31;
    const int lm   = lane & 15;
    const int koff = (lane >> 4) * 2;
    const bool hi  = (koff != 0);

    v8f c = {};
    for (int kk = 0; kk < N_NODES; kk += 4) {
        float4 af = *(const float4*)&A[(i + lm) * N_NODES + kk];
        if (kk + 64 < N_NODES)
            __builtin_prefetch(&A[(i + lm) * N_NODES + kk + 64], 0, 3);
        v2f a, b;
        a.x = hi ? af.z : af.x;
        a.y = hi ? af.w : af.y;
        b.x = v[(kk + koff)     * HID + j + lm];
        b.y = v[(kk + koff + 1) * HID + j + lm];
        c = __builtin_amdgcn_wmma_f32_16x16x4_f32(false, a, false, b, (short)0, c, false, false);
    }
    const int rbase = i + (lane >> 4) * 8;
#pragma unroll
    for (int r = 0; r < 8; ++r)
        M[(rbase + r) * HID + j + lm] = c[r];
}

// ---------------------------------------------------------------------------
// Kernel 6: per-edge post-softmax value scatter: M[s,h] += A[s,d] * rv_e[h]
// ---------------------------------------------------------------------------
__global__ __launch_bounds__(256) void edge_v_kernel(
    const int* __restrict__ eidx, const float* __restrict__ attr,
    const float* __restrict__ A, const float* __restrict__ Ev,
    const float* __restrict__ bias, const float* __restrict__ mult,
    float* __restrict__ M)
{
    const int wave = blockIdx.x * (blockDim.x >> 5) + (threadIdx.x >> 5);
    const int lane = threadIdx.x & 31;
    if (wave >= N_EDGES) return;

    const int s = eidx[wave];
    const int d = eidx[N_EDGES + wave];
    const float w0 = 1.0f / (1.0f + __expf(-(attr[wave] - bias[0]) * mult[0]));
    const float w1 = 1.0f - w0;
    const float a  = A[s * N_NODES + d];

#pragma unroll
    for (int t = 0; t < 2; ++t) {
        const int h = lane + t * 32;
        const float rv = w0 * Ev[h] + w1 * Ev[HID + h];
        atomicAdd(&M[s * HID + h], a * rv);
    }
}

// ---------------------------------------------------------------------------
extern "C" void kernel_launch(void* const* d_in, const int* in_sizes, int n_in,
                              void* d_out, int out_size, void* d_ws, size_t ws_size,
                              hipStream_t stream)
{
    const float* X    = (const float*)d_in[0];
    const int*   eidx = (const int*)  d_in[1];
    const float* attr = (const float*)d_in[2];
    const float* Wk   = (const float*)d_in[3];
    const float* Wq   = (const float*)d_in[4];
    const float* Wv   = (const float*)d_in[5];
    const float* Ek   = (const float*)d_in[6];
    const float* Eq   = (const float*)d_in[7];
    const float* Ev   = (const float*)d_in[8];
    const float* bias = (const float*)d_in[9];
    const float* mult = (const float*)d_in[10];
    float* out = (float*)d_out;

    // Workspace layout (floats): q | k | v | A   => ~5 MB total
    float* q = (float*)d_ws;
    float* k = q + N_NODES * HID;
    float* v = k + N_NODES * HID;
    float* A = v + N_NODES * HID;

    qkv_kernel   <<<dim3(HID/16, N_NODES/16, 3), 32, 0, stream>>>(X, Wk, Wq, Wv, q, k, v);
    scores_kernel<<<dim3(N_NODES/16, N_NODES/16), 32, 0, stream>>>(q, k, A);
    edge_qk_kernel<<<N_EDGES/8, 256, 0, stream>>>(eidx, attr, q, k, Ek, Eq, bias, mult, A);
    softmax_kernel<<<N_NODES, 256, 0, stream>>>(A);
    av_kernel    <<<dim3(HID/16, N_NODES/16), 32, 0, stream>>>(A, v, out);
    edge_v_kernel<<<N_EDGES/8, 256, 0, stream>>>(eidx, attr, A, Ev, bias, mult, out);
}